// NeuralLogicMachine_26938034881130
// MI455X (gfx1250) — compile-verified
//
#include <hip/hip_runtime.h>
#include <hip/hip_bf16.h>

typedef __attribute__((ext_vector_type(2))) float v2f;
typedef __attribute__((ext_vector_type(8))) float v8f;

#define CDIM 128
#define LDS_PITCH 132   // 132 % 64 == 4 -> conflict-free strided b64 reads; 132*4 % 16 == 0 -> float4 stores stay aligned

// ---------------------------------------------------------------------------
// softmax over T logits (T is tiny: 5). One thread.
// ---------------------------------------------------------------------------
__global__ void nlm_softmax_kernel(const float* __restrict__ a,
                                   float* __restrict__ w, int T) {
    if (blockIdx.x == 0 && threadIdx.x == 0) {
        float mx = -1e30f;
        for (int i = 0; i < T; ++i) mx = fmaxf(mx, a[i]);
        float s = 0.f;
        for (int i = 0; i < T; ++i) { float e = __expf(a[i] - mx); w[i] = e; s += e; }
        float inv = 1.0f / s;
        for (int i = 0; i < T; ++i) w[i] *= inv;
    }
}

// ---------------------------------------------------------------------------
// Out[N x 128] = X[N x 128] @ Wsub[128 x 128] + biasScale * bias[n]
// Block = 256 threads = 8 waves. Block owns row tile m0 = 16*blockIdx.x; the
// 16x128 X tile is staged once into LDS (shared by all 8 waves). Wave w owns
// column tile n0 = 16*w and runs 32x V_WMMA_F32_16X16X4_F32, with the A
// fragment fed by conflict-free ds_load_b64.
//
// f32 WMMA fragment layouts (ISA 7.12.2):
//   A 16x4:  lane%16 = M; VGPR j holds K = j + 2*(lane>=16)  -> float2/lane
//   B 4x16:  lane%16 = N; VGPR j holds K = j + 2*(lane>=16)  -> float2/lane
//   C 16x16: lane%16 = N; VGPR r holds M = r + 8*(lane>=16)  -> float8/lane
// ---------------------------------------------------------------------------
__global__ __launch_bounds__(256) void nlm_gemm_wmma_kernel(
    const float* __restrict__ X, const float* __restrict__ Wsub,
    const float* __restrict__ bias, float biasScale,
    float* __restrict__ Out, int Nn) {
    __shared__ float xt[16 * LDS_PITCH];

    const int tid  = threadIdx.x;
    const int lane = tid & 31;
    const int wave = tid >> 5;
    const int r16  = lane & 15;
    const int grp  = lane >> 4;            // 0: K={0,1}, 1: K={2,3}
    const int m0   = blockIdx.x * 16;
    const int n0   = wave * 16;

    // --- stage X[m0:m0+16, 0:128] into LDS: 2048 floats, 8 per thread ------
    {
        const int flat = tid * 8;          // two float4 groups per thread
        const int rr   = flat >> 7;        // row within tile (0..15)
        const int cc   = flat & 127;       // col (multiple of 8)
        int mrow = m0 + rr; if (mrow >= Nn) mrow = Nn - 1;   // clamp (keeps EXEC full)
        const float4* gsrc = reinterpret_cast<const float4*>(X + (size_t)mrow * CDIM + cc);
        float4* ldst = reinterpret_cast<float4*>(xt + rr * LDS_PITCH + cc);
        ldst[0] = gsrc[0];
        ldst[1] = gsrc[1];
    }
    __syncthreads();

    // --- 16x16 tile GEMM: K=128 in steps of 4 ------------------------------
    const float* __restrict__ arow = xt + r16 * LDS_PITCH;
    v8f acc = {};
    #pragma unroll 4
    for (int k0 = 0; k0 < CDIM; k0 += 4) {
        const int ka = k0 + 2 * grp;
        v2f afrag = *reinterpret_cast<const v2f*>(arow + ka);   // ds_load_b64, 8B-aligned
        v2f bfrag;
        bfrag.x = Wsub[(size_t)ka       * CDIM + n0 + r16];
        bfrag.y = Wsub[(size_t)(ka + 1) * CDIM + n0 + r16];
        acc = __builtin_amdgcn_wmma_f32_16x16x4_f32(
            /*neg_a=*/false, afrag, /*neg_b=*/false, bfrag,
            /*c_mod=*/(short)0, acc, /*reuse_a=*/false, /*reuse_b=*/false);
    }

    const float bv = bias[n0 + r16] * biasScale;   // branch-free bias

    // --- store: single whole-tile guard, unguarded fast path ---------------
    if (m0 + 16 <= Nn) {
        float* __restrict__ obase = Out + ((size_t)m0 + 8 * grp) * CDIM + n0 + r16;
        #pragma unroll
        for (int r = 0; r < 8; ++r)
            obase[(size_t)r * CDIM] = acc[r] + bv;
    } else {
        for (int r = 0; r < 8; ++r) {
            const int m = m0 + r + 8 * grp;
            if (m < Nn) Out[(size_t)m * CDIM + n0 + r16] = acc[r] + bv;
        }
    }
}

// ---------------------------------------------------------------------------
// counts[src[e]] += 1  (segment degree; float so finalize can divide directly)
// ---------------------------------------------------------------------------
__global__ __launch_bounds__(256) void nlm_degree_kernel(
    const int* __restrict__ src, float* __restrict__ counts, int E) {
    const int e = blockIdx.x * blockDim.x + threadIdx.x;
    if (e < E) atomicAdd(&counts[src[e]], 1.0f);
}

// ---------------------------------------------------------------------------
// agg[src[e], :] += relu(P[src[e], :] + Q[dst[e], :])
// One lane per (edge, 4 columns): idx>>5 = edge, (idx&31)*4 = column base.
// float4 loads of L2-resident P/Q rows; 4x global_atomic_add_f32 per lane.
// ---------------------------------------------------------------------------
__global__ __launch_bounds__(256) void nlm_edge_scatter_kernel(
    const int* __restrict__ src, const int* __restrict__ dst,
    const float* __restrict__ P, const float* __restrict__ Q,
    float* __restrict__ agg, int E) {
    const int idx = blockIdx.x * blockDim.x + threadIdx.x;
    const int e = idx >> 5;
    if (e >= E) return;
    const int l = idx & 31;
    const int s = src[e];
    const int d = dst[e];
    const float4 p = reinterpret_cast<const float4*>(P + (size_t)s * CDIM)[l];
    const float4 q = reinterpret_cast<const float4*>(Q + (size_t)d * CDIM)[l];
    float* __restrict__ ar = agg + (size_t)s * CDIM + l * 4;
    atomicAdd(ar + 0, fmaxf(p.x + q.x, 0.0f));
    atomicAdd(ar + 1, fmaxf(p.y + q.y, 0.0f));
    atomicAdd(ar + 2, fmaxf(p.z + q.z, 0.0f));
    atomicAdd(ar + 3, fmaxf(p.w + q.w, 0.0f));
}

// ---------------------------------------------------------------------------
// out[i] += w[t] * agg[i] / max(counts[i / 128], 1)
// ---------------------------------------------------------------------------
__global__ __launch_bounds__(256) void nlm_finalize_kernel(
    const float* __restrict__ agg, const float* __restrict__ counts,
    const float* __restrict__ w, int t, float* __restrict__ out, int total) {
    const int i = blockIdx.x * blockDim.x + threadIdx.x;
    if (i < total) {
        const float c = fmaxf(counts[i >> 7], 1.0f);
        out[i] += w[t] * agg[i] / c;
    }
}

// ---------------------------------------------------------------------------
// Host-side orchestration (graph-capture safe: only async ops on `stream`).
// ---------------------------------------------------------------------------
extern "C" void kernel_launch(void* const* d_in, const int* in_sizes, int n_in,
                              void* d_out, int out_size, void* d_ws, size_t ws_size,
                              hipStream_t stream) {
    const int Nn = in_sizes[0] / CDIM;          // 50000 nodes
    const int T  = in_sizes[3];                 // 5 edge types
    const int E  = in_sizes[4] / (2 * T);       // 200000 edges per type

    const float* x   = (const float*)d_in[0];   // [N, C]
    const float* W   = (const float*)d_in[1];   // [T, 2C, C]
    const float* b   = (const float*)d_in[2];   // [T, C]
    const float* att = (const float*)d_in[3];   // [T]
    const int*   ei  = (const int*)d_in[4];     // [T, 2, E]
    float* out = (float*)d_out;

    // workspace layout: P | Q | agg | counts | w   (256B-aligned slots)
    char* base = (char*)d_ws;
    const size_t szNC  = (size_t)Nn * CDIM * sizeof(float);
    const size_t slot  = (szNC + 255) & ~(size_t)255;
    const size_t szCnt = ((size_t)Nn * sizeof(float) + 255) & ~(size_t)255;
    float* P      = (float*)(base);
    float* Q      = (float*)(base + slot);
    float* agg    = (float*)(base + 2 * slot);
    float* counts = (float*)(base + 3 * slot);
    float* wv     = (float*)(base + 3 * slot + szCnt);
    (void)ws_size; (void)n_in;

    hipMemsetAsync(out, 0, (size_t)out_size * sizeof(float), stream);
    nlm_softmax_kernel<<<1, 1, 0, stream>>>(att, wv, T);

    const int mTiles     = (Nn + 15) / 16;
    const int degBlocks  = (E + 255) / 256;
    const int edgeBlocks = ((E * 32) + 255) / 256;
    const int finBlocks  = ((Nn * CDIM) + 255) / 256;

    for (int t = 0; t < T; ++t) {
        const float* Wt   = W + (size_t)t * 2 * CDIM * CDIM;   // rows 0..C-1   (x[src] half)
        const float* Wb   = Wt + (size_t)CDIM * CDIM;          // rows C..2C-1  (x[dst] half)
        const float* bt   = b + (size_t)t * CDIM;
        const int* src_t  = ei + (size_t)t * 2 * E;
        const int* dst_t  = src_t + E;

        hipMemsetAsync(agg, 0, szNC, stream);
        hipMemsetAsync(counts, 0, (size_t)Nn * sizeof(float), stream);

        nlm_gemm_wmma_kernel<<<mTiles, 256, 0, stream>>>(x, Wt, bt, 0.0f, P, Nn);
        nlm_gemm_wmma_kernel<<<mTiles, 256, 0, stream>>>(x, Wb, bt, 1.0f, Q, Nn);
        nlm_degree_kernel<<<degBlocks, 256, 0, stream>>>(src_t, counts, E);
        nlm_edge_scatter_kernel<<<edgeBlocks, 256, 0, stream>>>(src_t, dst_t, P, Q, agg, E);
        nlm_finalize_kernel<<<finBlocks, 256, 0, stream>>>(agg, counts, wv, t, out, Nn * CDIM);
    }
}